// SparseGridding_31413390803214
// MI455X (gfx1250) — compile-verified
//
#include <hip/hip_runtime.h>
#include <hip/hip_bf16.h>

// ---------------------------------------------------------------------------
// SparseGridding (voxel-grid clustering) for gfx1250.
// Dense key space (<= 8*50^3 = 1e6 < 2^20) -> flags + prefix-scan replaces
// jnp.unique; segment ops become L2-resident atomics (192MB L2 holds all
// scratch tables). Wave32/WGP-friendly block sizes (256 / 1024).
// ---------------------------------------------------------------------------

#define KS        (1 << 20)      // dense key space (power of two for the scan)
#define SCAN_T    256            // threads per scan block
#define SCAN_E    1024           // elements per scan block (4 per thread)
#define SCAN_NB   (KS / SCAN_E)  // = 1024 scan blocks
#define GRID_F    0.02f
#define I32MAX    2147483647

// stats layout: [0..2]=min coord xyz, [3..5]=max coord xyz, [6]=max batch,
//               [7]=num_unique
__device__ __forceinline__ long long voxel_key(const float* __restrict__ pos,
                                               const int* __restrict__ batch,
                                               int i, const int* __restrict__ stats) {
    int c0 = (int)(pos[3 * i + 0] / GRID_F) - stats[0];
    int c1 = (int)(pos[3 * i + 1] / GRID_F) - stats[1];
    int c2 = (int)(pos[3 * i + 2] / GRID_F) - stats[2];
    long long d0 = (long long)(stats[3] - stats[0] + 1);
    long long d1 = (long long)(stats[4] - stats[1] + 1);
    long long d2 = (long long)(stats[5] - stats[2] + 1);
    long long b  = (long long)batch[i];
    return ((b * d2 + (long long)c2) * d1 + (long long)c1) * d0 + (long long)c0;
}

// ---------------------------------------------------------------------------
__global__ void k_init(int* __restrict__ stats, int* __restrict__ flags,
                       int* __restrict__ rank, int* __restrict__ cnt,
                       int* __restrict__ minid, int* __restrict__ perm,
                       float* __restrict__ sum, float* __restrict__ sumoff) {
    int i = blockIdx.x * blockDim.x + threadIdx.x;   // covers 3*KS
    if (i < 3 * KS) { sum[i] = 0.0f; sumoff[i] = 0.0f; }
    if (i < KS) {
        flags[i] = 0; rank[i] = 0; cnt[i] = 0;
        minid[i] = I32MAX; perm[i] = 0;
    }
    if (i < 3)            stats[i] = I32MAX;          // mins
    else if (i < 7)       stats[i] = -I32MAX - 1;     // maxs + batch max
    else if (i == 7)      stats[i] = 0;               // num_unique
}

// ---------------------------------------------------------------------------
__global__ void k_minmax(const float* __restrict__ pos, const int* __restrict__ batch,
                         int npts, int* __restrict__ stats) {
    int lmin0 = I32MAX, lmin1 = I32MAX, lmin2 = I32MAX;
    int lmax0 = -I32MAX - 1, lmax1 = -I32MAX - 1, lmax2 = -I32MAX - 1;
    int lb = -I32MAX - 1;
    for (int i = blockIdx.x * blockDim.x + threadIdx.x; i < npts;
         i += gridDim.x * blockDim.x) {
        int c0 = (int)(pos[3 * i + 0] / GRID_F);
        int c1 = (int)(pos[3 * i + 1] / GRID_F);
        int c2 = (int)(pos[3 * i + 2] / GRID_F);
        lmin0 = min(lmin0, c0); lmax0 = max(lmax0, c0);
        lmin1 = min(lmin1, c1); lmax1 = max(lmax1, c1);
        lmin2 = min(lmin2, c2); lmax2 = max(lmax2, c2);
        lb    = max(lb, batch[i]);
    }
    __shared__ int sh[7][256];
    int t = threadIdx.x;
    sh[0][t] = lmin0; sh[1][t] = lmin1; sh[2][t] = lmin2;
    sh[3][t] = lmax0; sh[4][t] = lmax1; sh[5][t] = lmax2;
    sh[6][t] = lb;
    __syncthreads();
    for (int s = 128; s > 0; s >>= 1) {
        if (t < s) {
            sh[0][t] = min(sh[0][t], sh[0][t + s]);
            sh[1][t] = min(sh[1][t], sh[1][t + s]);
            sh[2][t] = min(sh[2][t], sh[2][t + s]);
            sh[3][t] = max(sh[3][t], sh[3][t + s]);
            sh[4][t] = max(sh[4][t], sh[4][t + s]);
            sh[5][t] = max(sh[5][t], sh[5][t + s]);
            sh[6][t] = max(sh[6][t], sh[6][t + s]);
        }
        __syncthreads();
    }
    if (t == 0) {
        atomicMin(&stats[0], sh[0][0]); atomicMin(&stats[1], sh[1][0]);
        atomicMin(&stats[2], sh[2][0]); atomicMax(&stats[3], sh[3][0]);
        atomicMax(&stats[4], sh[4][0]); atomicMax(&stats[5], sh[5][0]);
        atomicMax(&stats[6], sh[6][0]);
    }
}

// ---------------------------------------------------------------------------
__global__ void k_hist(const float* __restrict__ pos, const int* __restrict__ batch,
                       int npts, const int* __restrict__ stats,
                       int* __restrict__ flags) {
    int i = blockIdx.x * blockDim.x + threadIdx.x;
    if (i >= npts) return;
    long long key = voxel_key(pos, batch, i, stats);
    if (key >= 0 && key < (long long)KS) flags[key] = 1;
}

// ---------------------------------------------------------------------------
// Two-level exclusive prefix scan over the KS flag array (LDS Hillis-Steele).
__global__ void k_scanA(const int* __restrict__ flags, int* __restrict__ rank,
                        int* __restrict__ bsum) {
    __shared__ int sh[SCAN_T];
    int t = threadIdx.x;
    int base = blockIdx.x * SCAN_E + t * 4;
    int4 v = *(const int4*)(flags + base);
    int s1 = v.x, s2 = s1 + v.y, s3 = s2 + v.z, s4 = s3 + v.w;
    sh[t] = s4;
    __syncthreads();
    for (int o = 1; o < SCAN_T; o <<= 1) {
        int add = (t >= o) ? sh[t - o] : 0;
        __syncthreads();
        sh[t] += add;
        __syncthreads();
    }
    int excl = sh[t] - s4;
    int4 r; r.x = excl; r.y = excl + s1; r.z = excl + s2; r.w = excl + s3;
    *(int4*)(rank + base) = r;
    if (t == SCAN_T - 1) bsum[blockIdx.x] = sh[SCAN_T - 1];
}

__global__ void k_scanB(int* __restrict__ bsum, int* __restrict__ stats) {
    __shared__ int sh[SCAN_NB];
    int t = threadIdx.x;           // blockDim.x == SCAN_NB == 1024
    int v = bsum[t];
    sh[t] = v;
    __syncthreads();
    for (int o = 1; o < SCAN_NB; o <<= 1) {
        int add = (t >= o) ? sh[t - o] : 0;
        __syncthreads();
        sh[t] += add;
        __syncthreads();
    }
    bsum[t] = sh[t] - v;                       // exclusive
    if (t == SCAN_NB - 1) stats[7] = sh[SCAN_NB - 1];  // num_unique
}

__global__ void k_scanC(int* __restrict__ rank, const int* __restrict__ bsum) {
    int t = threadIdx.x;
    int base = blockIdx.x * SCAN_E + t * 4;
    int add = bsum[blockIdx.x];
    int4 v = *(int4*)(rank + base);
    v.x += add; v.y += add; v.z += add; v.w += add;
    *(int4*)(rank + base) = v;
}

// ---------------------------------------------------------------------------
__global__ void k_accum(const float* __restrict__ pos, const int* __restrict__ batch,
                        const int* __restrict__ ids, int npts,
                        const int* __restrict__ stats, const int* __restrict__ rank,
                        int* __restrict__ cnt, int* __restrict__ minid,
                        int* __restrict__ perm, float* __restrict__ sum) {
    int i = blockIdx.x * blockDim.x + threadIdx.x;
    if (i >= npts) return;
    long long key = voxel_key(pos, batch, i, stats);
    if (key < 0 || key >= (long long)KS) return;
    int s = rank[key];
    atomicAdd(&cnt[s], 1);
    atomicAdd(&sum[3 * s + 0], pos[3 * i + 0]);
    atomicAdd(&sum[3 * s + 1], pos[3 * i + 1]);
    atomicAdd(&sum[3 * s + 2], pos[3 * i + 2]);
    atomicMin(&minid[s], ids[i]);
    atomicMax(&perm[s], i);   // last occurrence == scatter-max of arange
}

// ---------------------------------------------------------------------------
__global__ void k_offsets(const float* __restrict__ pos, const int* __restrict__ batch,
                          int npts, const int* __restrict__ stats,
                          const int* __restrict__ rank, const int* __restrict__ cnt,
                          const float* __restrict__ sum, float* __restrict__ sumoff) {
    int i = blockIdx.x * blockDim.x + threadIdx.x;
    if (i >= npts) return;
    long long key = voxel_key(pos, batch, i, stats);
    if (key < 0 || key >= (long long)KS) return;
    int s = rank[key];
    float denom = fmaxf((float)cnt[s], 1.0f);
    #pragma unroll
    for (int k = 0; k < 3; ++k) {
        float mean = sum[3 * s + k] / denom;
        atomicAdd(&sumoff[3 * s + k], (pos[3 * i + k] - mean) / GRID_F);
    }
}

// ---------------------------------------------------------------------------
// Outputs (all float32, concatenated):
//   mean_pt[N,3] | x[N,3] | new_id[N] | new_batch[N] | new_coords[N,3]
__global__ void k_output(const float* __restrict__ pos, const int* __restrict__ batch,
                         int npts, const int* __restrict__ stats,
                         const int* __restrict__ cnt, const int* __restrict__ minid,
                         const int* __restrict__ perm, const float* __restrict__ sum,
                         const float* __restrict__ sumoff, float* __restrict__ out) {
    int s = blockIdx.x * blockDim.x + threadIdx.x;
    if (s >= npts) return;
    int nuniq = stats[7];
    size_t N = (size_t)npts;
    float m0, m1, m2, x0, x1, x2, fid, fb, cc0, cc1, cc2;
    if (s < nuniq) {
        float denom = fmaxf((float)cnt[s], 1.0f);
        m0 = sum[3 * s + 0] / denom; m1 = sum[3 * s + 1] / denom; m2 = sum[3 * s + 2] / denom;
        x0 = sumoff[3 * s + 0] / denom; x1 = sumoff[3 * s + 1] / denom; x2 = sumoff[3 * s + 2] / denom;
        fid = (float)minid[s];
        int p = perm[s];
        fb  = (float)batch[p];
        cc0 = (float)(int)(pos[3 * p + 0] / GRID_F);
        cc1 = (float)(int)(pos[3 * p + 1] / GRID_F);
        cc2 = (float)(int)(pos[3 * p + 2] / GRID_F);
    } else {
        // pad rows: count=0 -> mean=0, x=0; segment_min identity (int32 max);
        // perm=0 -> coords[0], batch[0]
        m0 = m1 = m2 = 0.0f; x0 = x1 = x2 = 0.0f;
        fid = 2147483648.0f;            // float(INT32_MAX)
        fb  = (float)batch[0];
        cc0 = (float)(int)(pos[0] / GRID_F);
        cc1 = (float)(int)(pos[1] / GRID_F);
        cc2 = (float)(int)(pos[2] / GRID_F);
    }
    out[3 * (size_t)s + 0] = m0;
    out[3 * (size_t)s + 1] = m1;
    out[3 * (size_t)s + 2] = m2;
    out[3 * N + 3 * (size_t)s + 0] = x0;
    out[3 * N + 3 * (size_t)s + 1] = x1;
    out[3 * N + 3 * (size_t)s + 2] = x2;
    out[6 * N + (size_t)s] = fid;
    out[7 * N + (size_t)s] = fb;
    out[8 * N + 3 * (size_t)s + 0] = cc0;
    out[8 * N + 3 * (size_t)s + 1] = cc1;
    out[8 * N + 3 * (size_t)s + 2] = cc2;
}

// ---------------------------------------------------------------------------
extern "C" void kernel_launch(void* const* d_in, const int* in_sizes, int n_in,
                              void* d_out, int out_size, void* d_ws, size_t ws_size,
                              hipStream_t stream) {
    const float* pos   = (const float*)d_in[0];
    const int*   batch = (const int*)d_in[1];
    const int*   ids   = (const int*)d_in[2];
    float*       out   = (float*)d_out;
    const int    npts  = in_sizes[1];        // batch element count == N

    // Workspace layout (all offsets 256B aligned; total ~44.0 MiB, L2-resident)
    char* ws = (char*)d_ws;
    size_t off = 0;
    int*   stats  = (int*)(ws + off);  off += 256;
    int*   flags  = (int*)(ws + off);  off += (size_t)KS * sizeof(int);
    int*   rank   = (int*)(ws + off);  off += (size_t)KS * sizeof(int);
    int*   bsum   = (int*)(ws + off);  off += (size_t)SCAN_NB * sizeof(int) + 256;
    int*   cnt    = (int*)(ws + off);  off += (size_t)KS * sizeof(int);
    int*   minid  = (int*)(ws + off);  off += (size_t)KS * sizeof(int);
    int*   perm   = (int*)(ws + off);  off += (size_t)KS * sizeof(int);
    float* sum    = (float*)(ws + off); off += (size_t)3 * KS * sizeof(float);
    float* sumoff = (float*)(ws + off); off += (size_t)3 * KS * sizeof(float);
    (void)ws_size; (void)n_in; (void)out_size; (void)ids;

    const int T  = 256;
    const int BN = (npts + T - 1) / T;          // point-parallel grids
    const int BI = (3 * KS + T - 1) / T;        // init grid

    k_init<<<BI, T, 0, stream>>>(stats, flags, rank, cnt, minid, perm, sum, sumoff);
    k_minmax<<<1024, T, 0, stream>>>(pos, batch, npts, stats);
    k_hist<<<BN, T, 0, stream>>>(pos, batch, npts, stats, flags);
    k_scanA<<<SCAN_NB, SCAN_T, 0, stream>>>(flags, rank, bsum);
    k_scanB<<<1, SCAN_NB, 0, stream>>>(bsum, stats);
    k_scanC<<<SCAN_NB, SCAN_T, 0, stream>>>(rank, bsum);
    k_accum<<<BN, T, 0, stream>>>(pos, batch, ids, npts, stats, rank, cnt, minid, perm, sum);
    k_offsets<<<BN, T, 0, stream>>>(pos, batch, npts, stats, rank, cnt, sum, sumoff);
    k_output<<<BN, T, 0, stream>>>(pos, batch, npts, stats, cnt, minid, perm, sum, sumoff, out);
}